// latentTransformationHead_72885595013444
// MI455X (gfx1250) — compile-verified
//
#include <hip/hip_runtime.h>
#include <stdint.h>

// ---------------------------------------------------------------------------
// The reference op is an identity on x (scatter-add of overlapping windows of
// x, normalized by overlap counts => c_j * x / c_j == x up to ~1 ulp).
// So the optimal MI455X kernel is a 256 MiB HBM-saturating copy (~11.5 us at
// 23.3 TB/s). We implement it with the CDNA5 Tensor Data Mover: each wave32
// issues TENSOR_LOAD_TO_LDS (32 KiB chunk) + TENSOR_STORE_FROM_LDS, tracked
// with TENSORcnt. DMA engines stream the data; SIMDs do almost nothing.
// ---------------------------------------------------------------------------

#if __has_include(<hip/amd_detail/amd_gfx1250_TDM.h>)
#define TDM_ARITY6 1   // amdgpu-toolchain (clang-23): 6-arg tensor builtins
#endif

typedef uint32_t u32x4 __attribute__((ext_vector_type(4)));
typedef int32_t  i32x4 __attribute__((ext_vector_type(4)));
typedef int32_t  i32x8 __attribute__((ext_vector_type(8)));

static constexpr uint32_t CHUNK_BYTES = 32768u;                  // 32 KiB tile per workgroup
static constexpr uint32_t ELEM_BYTES  = 8u;                      // D#.data_size code 3 => 8-byte elements
static constexpr uint32_t CHUNK_ELEMS = CHUNK_BYTES / ELEM_BYTES; // 4096 elements, fits tile_dim0 (16b)

// ---- D# group 0 (128b): count/valid, lds_addr, global_addr, type=2 --------
__device__ __forceinline__ u32x4 tdm_group0(uint32_t lds_addr, uint64_t gaddr) {
  u32x4 g;
  g[0] = 1u;                                   // count=1 (valid), is_restore=0, gather off
  g[1] = lds_addr;                             // LDS byte address of tile
  g[2] = (uint32_t)gaddr;                      // global_addr[31:0]
  g[3] = (uint32_t)((gaddr >> 32) & 0x01FFFFFFu) | (2u << 30); // global_addr[56:32] | type=2
  return g;
}

// ---- D# group 1 (256b): 1-D tensor of CHUNK_ELEMS 8-byte elements ---------
__device__ __forceinline__ i32x8 tdm_group1() {
  i32x8 g;
  g[0] = (int32_t)(3u << 16);                            // data_size=8B; wg_mask=0; no pad/iter/barrier
  g[1] = (int32_t)((CHUNK_ELEMS & 0xFFFFu) << 16);       // tensor_dim0[15:0] (hi half of dword)
  g[2] = (int32_t)((CHUNK_ELEMS >> 16) | (1u << 16));    // tensor_dim0[31:16] | tensor_dim1=1
  g[3] = (int32_t)((CHUNK_ELEMS & 0xFFFFu) << 16);       // tensor_dim1[31:16]=0 | tile_dim0=CHUNK_ELEMS
  g[4] = 0;                                              // tile_dim1=0 (unused), tile_dim2=0
  g[5] = (int32_t)CHUNK_ELEMS;                           // tensor_dim0_stride[31:0]
  g[6] = 0;                                              // stride0[47:32]=0 | stride1[15:0]=0
  g[7] = 0;                                              // stride1[47:16]=0
  return g;
}

// ---- D# groups 2/3: higher dims unused (dims=1, tiles=0) -------------------
__device__ __forceinline__ i32x4 tdm_group2() {
  i32x4 g; g[0] = 1; g[1] = 1; g[2] = 0; g[3] = 0;       // tensor_dim2=1, tensor_dim3=1
  return g;
}
__device__ __forceinline__ i32x4 tdm_group3() {
  i32x4 g; g[0] = 0; g[1] = (int32_t)(1u << 16); g[2] = 0; g[3] = 0; // tensor_dim4=1
  return g;
}

// One wave32 per block; each block DMAs one 32 KiB chunk: global->LDS->global.
__global__ __launch_bounds__(32) void tdm_copy_kernel(const float* __restrict__ src,
                                                      float* __restrict__ dst) {
  __shared__ __attribute__((aligned(128))) unsigned char lds_buf[CHUNK_BYTES];

  const uint64_t off = (uint64_t)blockIdx.x * (uint64_t)CHUNK_BYTES;
  const uint64_t sa  = (uint64_t)(uintptr_t)src + off;
  const uint64_t da  = (uint64_t)(uintptr_t)dst + off;
  // Flat LDS address: addr[31:0] is the byte offset within the workgroup's
  // LDS allocation (aperture bits live in addr[63:32]).
  const uint32_t lds = (uint32_t)(uintptr_t)&lds_buf[0];

  const u32x4 g0_load  = tdm_group0(lds, sa);
  const u32x4 g0_store = tdm_group0(lds, da);
  const i32x8 g1 = tdm_group1();
  const i32x4 g2 = tdm_group2();
  const i32x4 g3 = tdm_group3();

#ifdef TDM_ARITY6
  i32x8 g5 = {0, 0, 0, 0, 0, 0, 0, 0};
  __builtin_amdgcn_tensor_load_to_lds(g0_load, g1, g2, g3, g5, 0);
  __builtin_amdgcn_s_wait_tensorcnt(0);               // tile resident in LDS
  __builtin_amdgcn_tensor_store_from_lds(g0_store, g1, g2, g3, g5, 0);
#else
  __builtin_amdgcn_tensor_load_to_lds(g0_load, g1, g2, g3, 0);
  __builtin_amdgcn_s_wait_tensorcnt(0);               // tile resident in LDS
  __builtin_amdgcn_tensor_store_from_lds(g0_store, g1, g2, g3, 0);
#endif
  __builtin_amdgcn_s_wait_tensorcnt(0);               // store drained before wave exit
}

// Scalar tail for any bytes not covered by whole 32 KiB chunks (0 for the
// reference shape 8192x4096 f32, but keeps the launcher shape-generic).
__global__ void tail_copy_kernel(const float* __restrict__ src,
                                 float* __restrict__ dst, size_t n) {
  size_t i = (size_t)blockIdx.x * blockDim.x + threadIdx.x;
  if (i < n) dst[i] = src[i];
}

extern "C" void kernel_launch(void* const* d_in, const int* in_sizes, int n_in,
                              void* d_out, int out_size, void* d_ws, size_t ws_size,
                              hipStream_t stream) {
  (void)d_ws; (void)ws_size; (void)out_size;
  if (n_in < 1) return;

  const float* x   = (const float*)d_in[0];
  float*       out = (float*)d_out;

  const size_t n_floats = (size_t)in_sizes[0];
  const size_t n_bytes  = n_floats * sizeof(float);

  const size_t n_chunks     = n_bytes / CHUNK_BYTES;              // 4096 for 8192x4096 f32
  const size_t chunk_floats = n_chunks * (CHUNK_BYTES / sizeof(float));
  const size_t tail_floats  = n_floats - chunk_floats;            // 0 for reference shape

  if (n_chunks) {
    tdm_copy_kernel<<<dim3((uint32_t)n_chunks), dim3(32), 0, stream>>>(x, out);
  }
  if (tail_floats) {
    const uint32_t threads = 256;
    const uint32_t blocks  = (uint32_t)((tail_floats + threads - 1) / threads);
    tail_copy_kernel<<<dim3(blocks), dim3(threads), 0, stream>>>(
        x + chunk_floats, out + chunk_floats, tail_floats);
  }
}